// Separation_loss_20306605375876
// MI455X (gfx1250) — compile-verified
//
#include <hip/hip_runtime.h>
#include <hip/hip_bf16.h>

typedef float v2f __attribute__((ext_vector_type(2)));
typedef float v8f __attribute__((ext_vector_type(8)));

// ---- workspace layout (in floats), per scale ----
#define SCALE_STRIDE 512
#define OFF_MAX 0     // 16 x uint (encoded row max of normal)
#define OFF_S   16    // 16 x float (sum exp per row)
#define OFF_N2  32    // 16 x float (||normal_row||^2)
#define OFF_S2  48    // 16 x float (||noise_row||^2)
#define OFF_DOT 64    // 16 x float (noise.normal per row)
#define OFF_G   80    // 256 x float (Gram of exp(normal - max))

#define ENC_NEG_INF 0x007FFFFFu

__device__ __forceinline__ unsigned encF(float f) {
    unsigned u = __float_as_uint(f);
    return (u & 0x80000000u) ? ~u : (u | 0x80000000u);
}
__device__ __forceinline__ float decF(unsigned e) {
    unsigned u = (e & 0x80000000u) ? (e ^ 0x80000000u) : ~e;
    return __uint_as_float(u);
}

// ------------------------------------------------------------------
__global__ __launch_bounds__(256) void init_ws(float* ws) {
    int t = blockIdx.x * blockDim.x + threadIdx.x;
    if (t < 3 * SCALE_STRIDE) {
        int r = t % SCALE_STRIDE;
        if (r < 16) ((unsigned*)ws)[t] = ENC_NEG_INF;
        else        ws[t] = 0.0f;
    }
}

// ------------------------------------------------------------------
// Pass 1: per-row max of normal (for softmax), plus cosine-loss stats.
__global__ __launch_bounds__(256) void pass1_stats(const float* __restrict__ nrm,
                                                   const float* __restrict__ noi,
                                                   int D, float* __restrict__ wsBase) {
    __shared__ unsigned smMax[16];
    __shared__ float smN2[16], smS2[16], smDot[16];
    if (threadIdx.x < 16) {
        smMax[threadIdx.x] = ENC_NEG_INF;
        smN2[threadIdx.x] = 0.f; smS2[threadIdx.x] = 0.f; smDot[threadIdx.x] = 0.f;
    }
    __syncthreads();

    const int n4 = (D * 16) >> 2;
    const int stride = gridDim.x * blockDim.x;
    int curRow = -1;
    float lmax = -3.4e38f, ln2 = 0.f, ls2 = 0.f, ldot = 0.f;

    for (int i4 = blockIdx.x * blockDim.x + threadIdx.x; i4 < n4; i4 += stride) {
        int row = (i4 * 4) / D;   // D is a multiple of 4; float4 never spans rows
        if (row != curRow) {
            if (curRow >= 0) {
                atomicMax(&smMax[curRow], encF(lmax));
                atomicAdd(&smN2[curRow], ln2);
                atomicAdd(&smS2[curRow], ls2);
                atomicAdd(&smDot[curRow], ldot);
            }
            curRow = row; lmax = -3.4e38f; ln2 = 0.f; ls2 = 0.f; ldot = 0.f;
        }
        float4 a = ((const float4*)nrm)[i4];
        float4 b = ((const float4*)noi)[i4];
        lmax = fmaxf(lmax, fmaxf(fmaxf(a.x, a.y), fmaxf(a.z, a.w)));
        ln2  += a.x*a.x + a.y*a.y + a.z*a.z + a.w*a.w;
        ls2  += b.x*b.x + b.y*b.y + b.z*b.z + b.w*b.w;
        ldot += a.x*b.x + a.y*b.y + a.z*b.z + a.w*b.w;
    }
    if (curRow >= 0) {
        atomicMax(&smMax[curRow], encF(lmax));
        atomicAdd(&smN2[curRow], ln2);
        atomicAdd(&smS2[curRow], ls2);
        atomicAdd(&smDot[curRow], ldot);
    }
    __syncthreads();
    if (threadIdx.x < 16) {
        atomicMax(((unsigned*)(wsBase + OFF_MAX)) + threadIdx.x, smMax[threadIdx.x]);
        atomicAdd(wsBase + OFF_N2 + threadIdx.x, smN2[threadIdx.x]);
        atomicAdd(wsBase + OFF_S2 + threadIdx.x, smS2[threadIdx.x]);
        atomicAdd(wsBase + OFF_DOT + threadIdx.x, smDot[threadIdx.x]);
    }
}

// ------------------------------------------------------------------
// Pass 2: fused exp + row-sum + 16x16 Gram of E = exp(x - rowmax),
// accumulated in fp32 with V_WMMA_F32_16X16X4_F32 (A == B register pair).
// A layout (16x4 f32): lane L -> M = L&15, VGPR0 = K=2*(L>>4), VGPR1 = K+1.
// B layout (4x16 f32) for B = A^T is bit-identical, so we pass `a` twice.
__global__ __launch_bounds__(256) void pass2_gram(const float* __restrict__ x,
                                                  int D, float* __restrict__ wsBase) {
    const int lane  = threadIdx.x & 31;
    const int m     = lane & 15;
    const int khalf = lane >> 4;
    const float rowmax = decF(((const unsigned*)(wsBase + OFF_MAX))[m]);
    const float* rowp = x + (size_t)m * (size_t)D;

    const int wave   = (blockIdx.x * blockDim.x + threadIdx.x) >> 5;
    const int nwaves = (gridDim.x * blockDim.x) >> 5;

    v8f acc = {};
    float lsum = 0.0f;

    for (int k0 = wave * 64; k0 < D; k0 += nwaves * 64) {
#pragma unroll
        for (int s = 0; s < 16; ++s) {
            const int kb = k0 + 4 * s;
            const float2 v = *(const float2*)(rowp + kb + 2 * khalf);
            float e0 = __expf(v.x - rowmax);
            float e1 = __expf(v.y - rowmax);
            lsum += e0 + e1;
            v2f a; a.x = e0; a.y = e1;
            acc = __builtin_amdgcn_wmma_f32_16x16x4_f32(
                false, a, false, a, (short)0, acc, false, false);
        }
    }

    atomicAdd(wsBase + OFF_S + m, lsum);
    float* G = wsBase + OFF_G;
#pragma unroll
    for (int v = 0; v < 8; ++v) {
        int M = v + 8 * khalf;           // C/D layout: lanes 16-31 hold M+8
        atomicAdd(&G[M * 16 + m], acc[v]);
    }
}

// ------------------------------------------------------------------
// Final: normalize Gram -> P, build the four 16x16 cost matrices,
// run the 375-step annealed symmetric Sinkhorn, extrapolate at blur^2,
// add cosine-embedding loss, write scalar.
__global__ __launch_bounds__(1024) void sinkhorn_final(const float* __restrict__ ws,
                                                       const int* __restrict__ shuffle,
                                                       float* __restrict__ out) {
    const int tid = threadIdx.x;
    const int mat = tid >> 8;        // 0:f_ba 1:g_ab 2:f_aa 3:g_bb
    const int r   = tid & 255;
    const int i   = r >> 4;
    const int j   = r & 15;
    const int partner = (mat == 0) ? 1 : ((mat == 1) ? 0 : mat);

    __shared__ float P[16][16];
    __shared__ float phi[4][16];
    __shared__ int   sidx[16];
    __shared__ float red[64];
    __shared__ float cosred;

    const float a_log = -2.772588722f;   // -ln(16)
    float total = 0.0f;                  // meaningful in thread 0

    for (int sc = 0; sc < 3; ++sc) {
        const float* B = ws + sc * SCALE_STRIDE;
        __syncthreads();
        if (tid < 256) {
            int ii = tid >> 4, jj = tid & 15;
            P[ii][jj] = B[OFF_G + tid] / (B[OFF_S + ii] * B[OFF_S + jj]);
        }
        if (tid < 16) sidx[tid] = shuffle[tid];
        __syncthreads();

        int a, b;
        if      (mat == 0) { a = i;       b = sidx[j]; }   // C_xy
        else if (mat == 1) { a = sidx[i]; b = j;       }   // C_yx
        else if (mat == 2) { a = i;       b = j;       }   // C_xx
        else               { a = sidx[i]; b = sidx[j]; }   // C_yy
        const float Cv = 0.5f * fmaxf(P[a][a] + P[b][b] - 2.0f * P[a][b], 0.0f);

        // init: softmin(eps0=1e14, C, a_log)
        {
            float t = a_log - Cv * 1e-14f;
            float vmax = t;
            for (int o = 8; o >= 1; o >>= 1) vmax = fmaxf(vmax, __shfl_xor(vmax, o, 16));
            float e = __expf(t - vmax);
            for (int o = 8; o >= 1; o >>= 1) e += __shfl_xor(e, o, 16);
            float sm = -1e14f * (vmax + __logf(e));
            __syncthreads();
            if (j == 0) phi[mat][i] = sm;
            __syncthreads();
        }

        // eps schedule: [1e14, exp(32.23619 - 0.10258659*k) k=0..372, 0.0025]
        for (int it = 0; it < 375; ++it) {
            float eps;
            if (it == 0)        eps = 1e14f;
            else if (it == 374) eps = 0.0025f;
            else                eps = __expf(32.23619130f - 0.10258658877f * (float)(it - 1));
            const float ieps = 1.0f / eps;
            float own = phi[mat][i];
            float t = a_log + phi[partner][j] * ieps - Cv * ieps;
            float vmax = t;
            for (int o = 8; o >= 1; o >>= 1) vmax = fmaxf(vmax, __shfl_xor(vmax, o, 16));
            float e = __expf(t - vmax);
            for (int o = 8; o >= 1; o >>= 1) e += __shfl_xor(e, o, 16);
            float sm = -eps * (vmax + __logf(e));
            __syncthreads();
            if (j == 0) phi[mat][i] = 0.5f * (own + sm);
            __syncthreads();
        }

        // final extrapolation at eps = blur^2
        {
            const float eps = 0.0025f, ieps = 400.0f;
            float t = a_log + phi[partner][j] * ieps - Cv * ieps;
            float vmax = t;
            for (int o = 8; o >= 1; o >>= 1) vmax = fmaxf(vmax, __shfl_xor(vmax, o, 16));
            float e = __expf(t - vmax);
            for (int o = 8; o >= 1; o >>= 1) e += __shfl_xor(e, o, 16);
            float sm = -eps * (vmax + __logf(e));
            if (j == 0) red[mat * 16 + i] = ((mat < 2) ? sm : -sm) * 0.0625f;
            if (tid == 0) {
                float c = 0.0f;
                for (int bi = 0; bi < 16; ++bi) {
                    float nn = fmaxf(sqrtf(B[OFF_N2 + bi]), 1e-8f);
                    float ns = fmaxf(sqrtf(B[OFF_S2 + bi]), 1e-8f);
                    float cs = B[OFF_DOT + bi] / (ns * nn);
                    c += fmaxf(cs - 0.5f, 0.0f);
                }
                cosred = c * 0.0625f;
            }
            __syncthreads();
            if (tid == 0) {
                float s = 0.0f;
                for (int q = 0; q < 64; ++q) s += red[q];
                total += s + 0.1f * cosred;
            }
        }
    }
    __syncthreads();
    if (tid == 0) out[0] = total;
}

// ------------------------------------------------------------------
extern "C" void kernel_launch(void* const* d_in, const int* in_sizes, int n_in,
                              void* d_out, int out_size, void* d_ws, size_t ws_size,
                              hipStream_t stream) {
    // dict order: normal1, noise1, normal2, noise2, normal3, noise3, shuffle_index
    const float* normal[3] = { (const float*)d_in[0], (const float*)d_in[2], (const float*)d_in[4] };
    const float* noise [3] = { (const float*)d_in[1], (const float*)d_in[3], (const float*)d_in[5] };
    const int*   shuffle   = (const int*)d_in[6];
    float* ws  = (float*)d_ws;
    float* out = (float*)d_out;

    init_ws<<<6, 256, 0, stream>>>(ws);

    for (int s = 0; s < 3; ++s) {
        const int D  = in_sizes[2 * s] / 16;
        const int n4 = in_sizes[2 * s] / 4;
        int blocks1 = (n4 + 255) / 256; if (blocks1 > 2048) blocks1 = 2048;
        pass1_stats<<<blocks1, 256, 0, stream>>>(normal[s], noise[s], D, ws + s * SCALE_STRIDE);
    }
    for (int s = 0; s < 3; ++s) {
        const int D = in_sizes[2 * s] / 16;
        int chunks = D / 64;
        int waves  = chunks / 4; if (waves < 1) waves = 1;
        int blocks2 = (waves + 7) / 8;
        if (blocks2 > 2048) blocks2 = 2048;
        if (blocks2 < 1) blocks2 = 1;
        pass2_gram<<<blocks2, 256, 0, stream>>>(normal[s], D, ws + s * SCALE_STRIDE);
    }
    sinkhorn_final<<<1, 1024, 0, stream>>>(ws, shuffle, out);
}